// NeighborEmbedding_24824910971045
// MI455X (gfx1250) — compile-verified
//
#include <hip/hip_runtime.h>
#include <hip/hip_bf16.h>

// ---------------------------------------------------------------------------
// NeighborEmbedding for MI455X (gfx1250, wave32).
// f32 WMMA (V_WMMA_F32_16X16X4_F32) for all GEMM-shaped work; FPS/KNN in LDS.
// ---------------------------------------------------------------------------

typedef __attribute__((ext_vector_type(2))) float v2f;
typedef __attribute__((ext_vector_type(8))) float v8f;

namespace ne {
constexpr int B  = 16;
constexpr int N  = 2048;
constexpr int S0 = 256;
constexpr int S1 = 128;
constexpr int KN = 32;
constexpr float EPS = 1e-5f;
}

// ---------------------------------------------------------------- stage 1 MLP
// h1[b*N+n][o] = sum_c coords[b,n,c] * w1[o,c]; also writes coords transpose.
__global__ void __launch_bounds__(256) mlp1_kernel(
    const float* __restrict__ x, const float* __restrict__ w1,
    float* __restrict__ coords, float* __restrict__ h1) {
  int p = blockIdx.x * 256 + threadIdx.x;
  if (p >= ne::B * ne::N) return;
  int b = p / ne::N, n = p % ne::N;
  float c0 = x[(b * 3 + 0) * ne::N + n];
  float c1 = x[(b * 3 + 1) * ne::N + n];
  float c2 = x[(b * 3 + 2) * ne::N + n];
  coords[p * 3 + 0] = c0;
  coords[p * 3 + 1] = c1;
  coords[p * 3 + 2] = c2;
  #pragma unroll 8
  for (int o = 0; o < 64; ++o)
    h1[p * 64 + o] = c0 * w1[o * 3 + 0] + c1 * w1[o * 3 + 1] + c2 * w1[o * 3 + 2];
}

// ------------------------------------------------- BN stats over materialized h
// one block per channel; writes scale = gamma*rsqrt(var+eps), shift = beta-mean*scale
__global__ void __launch_bounds__(256) bn_stats_kernel(
    const float* __restrict__ h, int rows, int ch,
    const float* __restrict__ gamma, const float* __restrict__ beta,
    float* __restrict__ scale, float* __restrict__ shift) {
  __shared__ float ss[256], sq[256];
  int c = blockIdx.x, tid = threadIdx.x;
  float s1 = 0.f, s2 = 0.f;
  for (int r = tid; r < rows; r += 256) {
    float v = h[r * ch + c];
    s1 += v; s2 += v * v;
  }
  ss[tid] = s1; sq[tid] = s2;
  __syncthreads();
  for (int off = 128; off > 0; off >>= 1) {
    if (tid < off) { ss[tid] += ss[tid + off]; sq[tid] += sq[tid + off]; }
    __syncthreads();
  }
  if (tid == 0) {
    float mean = ss[0] / (float)rows;
    float var  = sq[0] / (float)rows - mean * mean;
    float inv  = rsqrtf(var + ne::EPS);
    float scl  = gamma[c] * inv;
    scale[c] = scl;
    shift[c] = beta[c] - mean * scl;
  }
}

// ------------------------------------------------- stage 2 MLP via f32 WMMA
// A = relu(h1*scale+shift) [rows,64]; h2 = A @ w2^T (raw, BN applied later)
__global__ void __launch_bounds__(256) mlp2_gemm_kernel(
    const float* __restrict__ h1, const float* __restrict__ scale,
    const float* __restrict__ shift, const float* __restrict__ w,
    float* __restrict__ h2) {
  constexpr int D = 64, LDA = D + 1;
  __shared__ float Atile[64 * LDA];
  int tid = threadIdx.x;
  int row0 = blockIdx.x * 64;
  for (int e = tid; e < 64 * D; e += 256) {
    int r = e / D, d = e % D;
    float v = h1[(row0 + r) * D + d] * scale[d] + shift[d];
    Atile[r * LDA + d] = fmaxf(v, 0.f);
  }
  __syncthreads();
  int wave = tid >> 5, lane = tid & 31;
  constexpr int TN = D / 16, TT = 4 * TN;   // 16 tiles, 2 per wave
  for (int t = wave; t < TT; t += 8) {
    int mtile = t / TN, ntile = t % TN;
    int mrow = mtile * 16 + (lane & 15);
    int ncol = ntile * 16 + (lane & 15);
    int kh = (lane >> 4) * 2;
    v8f acc = {};
    for (int kk = 0; kk < D; kk += 4) {
      v2f a, bb;
      a[0]  = Atile[mrow * LDA + kk + kh];
      a[1]  = Atile[mrow * LDA + kk + kh + 1];
      bb[0] = w[ncol * D + kk + kh];
      bb[1] = w[ncol * D + kk + kh + 1];
      acc = __builtin_amdgcn_wmma_f32_16x16x4_f32(false, a, false, bb,
                                                  (short)0, acc, false, false);
    }
    int rbase = mtile * 16 + ((lane >> 4) << 3);
    #pragma unroll
    for (int r = 0; r < 8; ++r)
      h2[(row0 + rbase + r) * D + ncol] = acc[r];
  }
}

// ------------------------------------------------------------ BN+ReLU apply
__global__ void __launch_bounds__(256) apply_bn_relu_kernel(
    const float* __restrict__ h, const float* __restrict__ scale,
    const float* __restrict__ shift, float* __restrict__ o, int total, int ch) {
  int i = blockIdx.x * 256 + threadIdx.x;
  if (i >= total) return;
  int c = i % ch;
  o[i] = fmaxf(h[i] * scale[c] + shift[c], 0.f);
}

// ------------------------------------------------------------------ FPS scan
// one workgroup per batch; sequential S-step scan, dist in LDS, tree argmax.
__global__ void __launch_bounds__(256) fps_kernel(
    const float* __restrict__ pts, int NP, int S,
    int* __restrict__ fps_idx, float* __restrict__ new_xyz) {
  __shared__ float dist[2048];
  __shared__ float rval[256];
  __shared__ int   ridx[256];
  __shared__ float cen[3];
  __shared__ int   curfar;
  int b = blockIdx.x, tid = threadIdx.x;
  const float* P = pts + (size_t)b * NP * 3;
  for (int i = tid; i < NP; i += 256) dist[i] = 1e10f;
  if (tid == 0) curfar = 0;
  __syncthreads();
  for (int s = 0; s < S; ++s) {
    if (tid == 0) {
      int far = curfar;
      fps_idx[b * S + s] = far;
      cen[0] = P[far * 3 + 0];
      cen[1] = P[far * 3 + 1];
      cen[2] = P[far * 3 + 2];
      new_xyz[(b * S + s) * 3 + 0] = cen[0];
      new_xyz[(b * S + s) * 3 + 1] = cen[1];
      new_xyz[(b * S + s) * 3 + 2] = cen[2];
    }
    __syncthreads();
    float best = -1.f; int bi = 0;
    for (int i = tid; i < NP; i += 256) {
      float dx = P[i * 3 + 0] - cen[0];
      float dy = P[i * 3 + 1] - cen[1];
      float dz = P[i * 3 + 2] - cen[2];
      float d = fminf(dist[i], dx * dx + dy * dy + dz * dz);
      dist[i] = d;
      if (d > best) { best = d; bi = i; }
    }
    rval[tid] = best; ridx[tid] = bi;
    __syncthreads();
    for (int off = 128; off > 0; off >>= 1) {
      if (tid < off) {
        float ov = rval[tid + off]; int oi = ridx[tid + off];
        if (ov > rval[tid] || (ov == rval[tid] && oi < ridx[tid])) {
          rval[tid] = ov; ridx[tid] = oi;
        }
      }
      __syncthreads();
    }
    if (tid == 0) curfar = ridx[0];
    __syncthreads();
  }
}

// ---------------------------------------------------------------------- KNN
// one workgroup per (b,s) center; 32 iterative argmins over LDS d2 array.
__global__ void __launch_bounds__(256) knn_kernel(
    const float* __restrict__ centers, const float* __restrict__ pts,
    int NP, int S, int* __restrict__ knn) {
  __shared__ float d2[2048];
  __shared__ float rval[256];
  __shared__ int   ridx[256];
  int bs = blockIdx.x, tid = threadIdx.x;
  int b = bs / S;
  float cx = centers[bs * 3 + 0];
  float cy = centers[bs * 3 + 1];
  float cz = centers[bs * 3 + 2];
  const float* P = pts + (size_t)b * NP * 3;
  for (int i = tid; i < NP; i += 256) {
    float dx = P[i * 3 + 0] - cx;
    float dy = P[i * 3 + 1] - cy;
    float dz = P[i * 3 + 2] - cz;
    d2[i] = dx * dx + dy * dy + dz * dz;
  }
  __syncthreads();
  for (int kk = 0; kk < ne::KN; ++kk) {
    float best = 1e30f; int bi = 0x7fffffff;
    for (int i = tid; i < NP; i += 256) {
      float v = d2[i];
      if (v < best || (v == best && i < bi)) { best = v; bi = i; }
    }
    rval[tid] = best; ridx[tid] = bi;
    __syncthreads();
    for (int off = 128; off > 0; off >>= 1) {
      if (tid < off) {
        float ov = rval[tid + off]; int oi = ridx[tid + off];
        if (ov < rval[tid] || (ov == rval[tid] && oi < ridx[tid])) {
          rval[tid] = ov; ridx[tid] = oi;
        }
      }
      __syncthreads();
    }
    if (tid == 0) { knn[bs * ne::KN + kk] = ridx[0]; d2[ridx[0]] = 1e30f; }
    __syncthreads();
  }
}

// --------------------------------------------- SG gather-GEMM (f32 WMMA core)
// One workgroup = one (b,s) group of KN=32 rows, D=2C columns in/out.
// pass 0: accumulate per-channel sum/sumsq of h=nf@w^T into accum[2D] (atomics)
// pass 1: h -> BN+ReLU -> max over k -> out ([B,S,D] or transposed [B,D,S]).
template <int C>
__global__ void __launch_bounds__(256) sg_gemm_kernel(
    const float* __restrict__ feat, int NP,
    const int* __restrict__ fps_idx, const int* __restrict__ knn_idx, int S,
    const float* __restrict__ w, int pass,
    float* __restrict__ accum,
    const float* __restrict__ scale, const float* __restrict__ shift,
    float* __restrict__ outp, int out_transposed) {
  constexpr int D = 2 * C, LDA = D + 1;
  constexpr int TN = D / 16, TT = 2 * TN, TPW = TT / 8;
  __shared__ float Atile[32 * LDA];   // reused as H tile in pass 1 (phase 2)
  __shared__ int   nbbase[32], ctbase[32];
  __shared__ float colsum[D], colsq[D];

  int tid = threadIdx.x;
  int bs = blockIdx.x;                // == b*S + s (32 rows == one knn group)
  if (tid < 32) {
    int r = bs * 32 + tid;
    int b = r / (S * ne::KN);
    int s = (r / ne::KN) % S;
    int k = r % ne::KN;
    nbbase[tid] = (b * NP + knn_idx[(b * S + s) * ne::KN + k]) * C;
    ctbase[tid] = (b * NP + fps_idx[b * S + s]) * C;
  }
  for (int i = tid; i < D; i += 256) { colsum[i] = 0.f; colsq[i] = 0.f; }
  __syncthreads();

  // gather concat( grouped - center , center ) tile into LDS
  for (int e = tid; e < 32 * D; e += 256) {
    int r = e / D, d = e % D;
    float v = (d < C) ? (feat[nbbase[r] + d] - feat[ctbase[r] + d])
                      : feat[ctbase[r] + d - C];
    Atile[r * LDA + d] = v;
  }
  __syncthreads();

  int wave = tid >> 5, lane = tid & 31;
  int kh = (lane >> 4) * 2;
  v8f accs[TPW];

  // phase 1: all WMMA tiles, accumulators kept in registers
  for (int i = 0; i < TPW; ++i) {
    int t = wave + i * 8;
    int mrow = (t / TN) * 16 + (lane & 15);
    int ncol = (t % TN) * 16 + (lane & 15);
    __builtin_prefetch(&w[ncol * D], 0, 3);
    v8f acc = {};
    for (int kk = 0; kk < D; kk += 4) {
      v2f a, bb;
      a[0]  = Atile[mrow * LDA + kk + kh];
      a[1]  = Atile[mrow * LDA + kk + kh + 1];
      bb[0] = w[ncol * D + kk + kh];
      bb[1] = w[ncol * D + kk + kh + 1];
      acc = __builtin_amdgcn_wmma_f32_16x16x4_f32(false, a, false, bb,
                                                  (short)0, acc, false, false);
    }
    accs[i] = acc;
  }
  __syncthreads();   // A tile fully consumed -> safe to reuse for H

  if (pass == 0) {
    for (int i = 0; i < TPW; ++i) {
      int t = wave + i * 8;
      int ncol = (t % TN) * 16 + (lane & 15);
      float s1 = 0.f, s2 = 0.f;
      #pragma unroll
      for (int r = 0; r < 8; ++r) { float v = accs[i][r]; s1 += v; s2 += v * v; }
      atomicAdd(&colsum[ncol], s1);
      atomicAdd(&colsq[ncol], s2);
    }
    __syncthreads();
    for (int i = tid; i < D; i += 256) {
      atomicAdd(&accum[i],     colsum[i]);
      atomicAdd(&accum[D + i], colsq[i]);
    }
  } else {
    for (int i = 0; i < TPW; ++i) {
      int t = wave + i * 8;
      int ncol  = (t % TN) * 16 + (lane & 15);
      int rbase = (t / TN) * 16 + ((lane >> 4) << 3);
      float sc = scale[ncol], sh = shift[ncol];
      #pragma unroll
      for (int r = 0; r < 8; ++r)
        Atile[(rbase + r) * LDA + ncol] = fmaxf(accs[i][r] * sc + sh, 0.f);
    }
    __syncthreads();
    // max-pool the 32 rows -> one output row (post-ReLU values are >= 0)
    for (int col = tid; col < D; col += 256) {
      float m = 0.f;
      #pragma unroll 8
      for (int r = 0; r < 32; ++r) m = fmaxf(m, Atile[r * LDA + col]);
      if (out_transposed) {
        int b = bs / S, s = bs % S;
        outp[(b * D + col) * S + s] = m;
      } else {
        outp[bs * D + col] = m;
      }
    }
  }
}

// ----------------------------------------------------- finalize SG BN params
__global__ void bn_finalize_kernel(const float* __restrict__ accum, float count,
                                   int ch, const float* __restrict__ gamma,
                                   const float* __restrict__ beta,
                                   float* __restrict__ scale,
                                   float* __restrict__ shift) {
  int c = blockIdx.x * blockDim.x + threadIdx.x;
  if (c >= ch) return;
  float mean = accum[c] / count;
  float var  = accum[ch + c] / count - mean * mean;
  float inv  = rsqrtf(var + ne::EPS);
  float scl  = gamma[c] * inv;
  scale[c] = scl;
  shift[c] = beta[c] - mean * scl;
}

__global__ void zero_kernel(float* __restrict__ p, int n) {
  int i = blockIdx.x * blockDim.x + threadIdx.x;
  if (i < n) p[i] = 0.f;
}

// ---------------------------------------------------------------------------
extern "C" void kernel_launch(void* const* d_in, const int* in_sizes, int n_in,
                              void* d_out, int out_size, void* d_ws, size_t ws_size,
                              hipStream_t stream) {
  using namespace ne;
  const float* x     = (const float*)d_in[0];
  const float* w1    = (const float*)d_in[1];
  const float* g1    = (const float*)d_in[2];
  const float* b1    = (const float*)d_in[3];
  const float* w2    = (const float*)d_in[4];
  const float* g2    = (const float*)d_in[5];
  const float* b2    = (const float*)d_in[6];
  const float* w_sg0 = (const float*)d_in[7];
  const float* g_sg0 = (const float*)d_in[8];
  const float* b_sg0 = (const float*)d_in[9];
  const float* w_sg1 = (const float*)d_in[10];
  const float* g_sg1 = (const float*)d_in[11];
  const float* b_sg1 = (const float*)d_in[12];
  float* out = (float*)d_out;

  // ---- workspace carve (≈28 MB, all re-initialized every call) ----
  char* p = (char*)d_ws;
  auto carve = [&](size_t elems) {
    void* r = (void*)p;
    p += ((elems * 4 + 255) / 256) * 256;
    return r;
  };
  float* coords = (float*)carve((size_t)B * N * 3);
  float* h1raw  = (float*)carve((size_t)B * N * 64);
  float* h2raw  = (float*)carve((size_t)B * N * 64);
  float* feat   = (float*)carve((size_t)B * N * 64);
  float* f1     = (float*)carve((size_t)B * S0 * 128);
  float* sc1    = (float*)carve(64);
  float* sh1    = (float*)carve(64);
  float* sc2    = (float*)carve(64);
  float* sh2    = (float*)carve(64);
  float* xyz0   = (float*)carve((size_t)B * S0 * 3);
  float* xyz1   = (float*)carve((size_t)B * S1 * 3);
  int*   fps0   = (int*)carve((size_t)B * S0);
  int*   fps1   = (int*)carve((size_t)B * S1);
  int*   knn0   = (int*)carve((size_t)B * S0 * KN);
  int*   knn1   = (int*)carve((size_t)B * S1 * KN);
  float* accum0 = (float*)carve(2 * 128);
  float* accum1 = (float*)carve(2 * 256);
  float* scg0   = (float*)carve(128);
  float* shg0   = (float*)carve(128);
  float* scg1   = (float*)carve(256);
  float* shg1   = (float*)carve(256);

  // ---- stage 1: per-point MLPs with batch-stat BN ----
  mlp1_kernel<<<(B * N + 255) / 256, 256, 0, stream>>>(x, w1, coords, h1raw);
  bn_stats_kernel<<<64, 256, 0, stream>>>(h1raw, B * N, 64, g1, b1, sc1, sh1);
  mlp2_gemm_kernel<<<(B * N) / 64, 256, 0, stream>>>(h1raw, sc1, sh1, w2, h2raw);
  bn_stats_kernel<<<64, 256, 0, stream>>>(h2raw, B * N, 64, g2, b2, sc2, sh2);
  apply_bn_relu_kernel<<<(B * N * 64 + 255) / 256, 256, 0, stream>>>(
      h2raw, sc2, sh2, feat, B * N * 64, 64);

  // ---- SG block 0: FPS(256) + KNN(32) + gather-GEMM(128ch) ----
  fps_kernel<<<B, 256, 0, stream>>>(coords, N, S0, fps0, xyz0);
  knn_kernel<<<B * S0, 256, 0, stream>>>(xyz0, coords, N, S0, knn0);
  zero_kernel<<<3, 256, 0, stream>>>(accum0, 2 * 128 + 2 * 256);  // accum0+accum1 contiguous
  sg_gemm_kernel<64><<<B * S0, 256, 0, stream>>>(
      feat, N, fps0, knn0, S0, w_sg0, /*pass=*/0, accum0, nullptr, nullptr,
      nullptr, 0);
  bn_finalize_kernel<<<1, 128, 0, stream>>>(accum0, (float)(B * S0 * KN), 128,
                                            g_sg0, b_sg0, scg0, shg0);
  sg_gemm_kernel<64><<<B * S0, 256, 0, stream>>>(
      feat, N, fps0, knn0, S0, w_sg0, /*pass=*/1, nullptr, scg0, shg0, f1, 0);

  // ---- SG block 1: FPS(128) + KNN(32) + gather-GEMM(256ch), transposed out --
  fps_kernel<<<B, 256, 0, stream>>>(xyz0, S0, S1, fps1, xyz1);
  knn_kernel<<<B * S1, 256, 0, stream>>>(xyz1, xyz0, S0, S1, knn1);
  sg_gemm_kernel<128><<<B * S1, 256, 0, stream>>>(
      f1, S0, fps1, knn1, S1, w_sg1, /*pass=*/0, accum1, nullptr, nullptr,
      nullptr, 0);
  bn_finalize_kernel<<<1, 256, 0, stream>>>(accum1, (float)(B * S1 * KN), 256,
                                            g_sg1, b_sg1, scg1, shg1);
  sg_gemm_kernel<128><<<B * S1, 256, 0, stream>>>(
      f1, S0, fps1, knn1, S1, w_sg1, /*pass=*/1, nullptr, scg1, shg1, out, 1);
}